// GraphEncoder_28879360098474
// MI455X (gfx1250) — compile-verified
//
#include <hip/hip_runtime.h>
#include <math.h>

// ---------------------------------------------------------------------------
// GCN encoder for MI455X (gfx1250, wave32).
//   - FP32 WMMA (v_wmma_f32_16x16x4_f32) for the two 100000x128x128 GEMMs.
//   - Wave-per-edge scatter with float4 loads + global_atomic_add_f32.
//   - Wave-per-row LayerNorm with wave32 butterfly reductions.
// ---------------------------------------------------------------------------

typedef float v2f __attribute__((ext_vector_type(2)));
typedef float v8f __attribute__((ext_vector_type(8)));

#define FD 128          // feature dim (D == H == O == 128)
#define EPS_LN 1e-5f
#define EPS_NORM 1e-12f

__device__ __forceinline__ float wave_sum(float v) {
#pragma unroll
    for (int off = 16; off > 0; off >>= 1)
        v += __shfl_xor(v, off, 32);
    return v;
}

// K1: per-edge sigmoid weight, raw-weight degree accumulation, global sum.
__global__ void __launch_bounds__(256)
edge_sig_deg(const float* __restrict__ ea, const int* __restrict__ dst,
             const float* __restrict__ dtw, const float* __restrict__ dtb,
             float* __restrict__ ew, float* __restrict__ deg,
             float* __restrict__ sum, int E)
{
    int i = blockIdx.x * blockDim.x + threadIdx.x;
    float s = 0.0f;
    if (i < E) {
        float z = fmaf(ea[i], dtw[0], dtb[0]);
        s = 1.0f / (1.0f + __expf(-z));       // sigmoid (positive -> |.| == .)
        ew[i] = s;
        atomicAdd(&deg[dst[i]], s);           // raw degree; 1/sum folded later
    }
    float ws = wave_sum(s);
    if ((threadIdx.x & 31) == 0) atomicAdd(sum, ws);
}

// K2: dinv[i] = rsqrt(deg_raw[i] * inv_sum + 1)   (self-loop weight 1)
__global__ void __launch_bounds__(256)
dinv_kernel(const float* __restrict__ deg, const float* __restrict__ sum,
            float* __restrict__ dinv, int n)
{
    int i = blockIdx.x * blockDim.x + threadIdx.x;
    if (i < n) {
        float inv = 1.0f / fmaxf(sum[0], EPS_NORM);
        dinv[i] = rsqrtf(fmaf(deg[i], inv, 1.0f));
    }
}

// K3: H = A(Mx128) @ W(128x128) with V_WMMA_F32_16X16X4_F32.
// One wave -> 16 rows x 64 cols (4 accumulators), K stepped by 4.
__global__ void __launch_bounds__(128)
gemm128_wmma(const float* __restrict__ A, const float* __restrict__ W,
             float* __restrict__ Hout, int M)
{
    int wave = (blockIdx.x * blockDim.x + threadIdx.x) >> 5;
    int lane = threadIdx.x & 31;
    int mTile = wave >> 1;                 // 16-row tile index
    int n0    = (wave & 1) * 64;           // column half
    int m0    = mTile * 16;
    if (m0 >= M) return;
    bool hi = lane >= 16;
    int  l  = lane & 15;

    const float* arow = A + (size_t)(m0 + l) * FD;   // A row for this lane
    v8f c0 = {}, c1 = {}, c2 = {}, c3 = {};

#pragma unroll 4
    for (int k = 0; k < FD; k += 4) {
        int ks = k + (hi ? 2 : 0);
        // A fragment (16x4 f32): lanes 0-15 K={k,k+1}, lanes 16-31 K={k+2,k+3}
        v2f a;
        a.x = arow[ks];
        a.y = arow[ks + 1];
        // B fragments (4x16 f32), mirrored layout over N
        const float* wb = W + (size_t)ks * FD + n0 + l;
        v2f b0; b0.x = wb[0];  b0.y = wb[FD];
        v2f b1; b1.x = wb[16]; b1.y = wb[FD + 16];
        v2f b2; b2.x = wb[32]; b2.y = wb[FD + 32];
        v2f b3; b3.x = wb[48]; b3.y = wb[FD + 48];
        c0 = __builtin_amdgcn_wmma_f32_16x16x4_f32(false, a, false, b0, (short)0, c0, false, false);
        c1 = __builtin_amdgcn_wmma_f32_16x16x4_f32(false, a, false, b1, (short)0, c1, false, false);
        c2 = __builtin_amdgcn_wmma_f32_16x16x4_f32(false, a, false, b2, (short)0, c2, false, false);
        c3 = __builtin_amdgcn_wmma_f32_16x16x4_f32(false, a, false, b3, (short)0, c3, false, false);
    }

    // C/D layout: VGPR v -> row m0 + v (+8 for hi half-wave), col n0 + 16j + l
    int rbase = m0 + (hi ? 8 : 0);
#pragma unroll
    for (int v = 0; v < 8; ++v) {
        float* o = Hout + (size_t)(rbase + v) * FD + n0 + l;
        o[0]  = c0[v];
        o[16] = c1[v];
        o[32] = c2[v];
        o[48] = c3[v];
    }
}

// K4: wave-per-edge scatter-add: agg[dst] += h[src] * (ew*inv_sum*dinv[s]*dinv[d])
__global__ void __launch_bounds__(256)
scatter_edges(const float* __restrict__ h, const float* __restrict__ ew,
              const float* __restrict__ sum, const float* __restrict__ dinv,
              const int* __restrict__ src, const int* __restrict__ dst,
              float* __restrict__ agg, int E)
{
    int wave = (blockIdx.x * blockDim.x + threadIdx.x) >> 5;
    int lane = threadIdx.x & 31;
    if (wave >= E) return;
    int s = src[wave];
    int d = dst[wave];
    float inv  = 1.0f / fmaxf(sum[0], EPS_NORM);
    float coef = ew[wave] * inv * dinv[s] * dinv[d];
    const float4* hs = (const float4*)(h + (size_t)s * FD);
    float4 v = hs[lane];                       // B128 coalesced row load
    float* ad = agg + (size_t)d * FD + lane * 4;
    atomicAdd(ad + 0, v.x * coef);
    atomicAdd(ad + 1, v.y * coef);
    atomicAdd(ad + 2, v.z * coef);
    atomicAdd(ad + 3, v.w * coef);
}

// K5: out = LN(agg + h*dinv^2 + bias) * g + be  (optional ReLU), wave per row.
template <bool RELU>
__global__ void __launch_bounds__(256)
finish_ln(const float* __restrict__ agg, const float* __restrict__ h,
          const float* __restrict__ dinv, const float* __restrict__ bias,
          const float* __restrict__ g, const float* __restrict__ be,
          float* __restrict__ out, int n)
{
    int row  = (blockIdx.x * blockDim.x + threadIdx.x) >> 5;
    int lane = threadIdx.x & 31;
    if (row >= n) return;
    float dv = dinv[row];
    float sl = dv * dv;
    const float* ar = agg + (size_t)row * FD;
    const float* hr = h + (size_t)row * FD;
    float v[4];
    float s = 0.0f;
#pragma unroll
    for (int j = 0; j < 4; ++j) {
        int dd = lane + 32 * j;
        v[j] = fmaf(hr[dd], sl, ar[dd]) + bias[dd];
        s += v[j];
    }
    float mean = wave_sum(s) * (1.0f / FD);
    float vs = 0.0f;
#pragma unroll
    for (int j = 0; j < 4; ++j) {
        float t = v[j] - mean;
        vs += t * t;
    }
    float var = wave_sum(vs) * (1.0f / FD);
    float r = rsqrtf(var + EPS_LN);
#pragma unroll
    for (int j = 0; j < 4; ++j) {
        int dd = lane + 32 * j;
        float y = (v[j] - mean) * r * g[dd] + be[dd];
        if (RELU) y = fmaxf(y, 0.0f);
        out[(size_t)row * FD + dd] = y;
    }
}

extern "C" void kernel_launch(void* const* d_in, const int* in_sizes, int n_in,
                              void* d_out, int out_size, void* d_ws, size_t ws_size,
                              hipStream_t stream)
{
    const float* x    = (const float*)d_in[0];
    const int*   ei   = (const int*)  d_in[1];
    const float* ea   = (const float*)d_in[2];
    const float* dt_w = (const float*)d_in[3];
    const float* dt_b = (const float*)d_in[4];
    const float* W1   = (const float*)d_in[5];
    const float* b1   = (const float*)d_in[6];
    const float* g1   = (const float*)d_in[7];
    const float* be1  = (const float*)d_in[8];
    const float* W2   = (const float*)d_in[9];
    const float* b2   = (const float*)d_in[10];
    const float* g2   = (const float*)d_in[11];
    const float* be2  = (const float*)d_in[12];
    float* out = (float*)d_out;

    const int n = in_sizes[0] / FD;     // 100000
    const int E = in_sizes[2];          // 1600000
    const int* src = ei;
    const int* dst = ei + E;

    // Workspace layout (floats): ew[E] | deg[n] | dinv[n] | sum[16] | bufA[n*FD] | bufB[n*FD]
    float* ws   = (float*)d_ws;
    float* ew   = ws;
    float* deg  = ew + (size_t)E;
    float* dinv = deg + (size_t)n;
    float* sum  = dinv + (size_t)n;
    float* bufA = sum + 16;
    float* bufB = bufA + (size_t)n * FD;

    const int mtiles = (n + 15) / 16;
    const int gwaves = mtiles * 2;                  // 16x64 slabs
    const int gemmBlocks = (gwaves + 3) / 4;        // 4 waves / block
    const int edgeBlocks = (E + 7) / 8;             // 8 edge-waves / block
    const int rowBlocks  = (n + 7) / 8;             // 8 row-waves / block

    // ---- edge weights + degrees -------------------------------------------
    hipMemsetAsync(deg, 0, sizeof(float) * (size_t)n, stream);
    hipMemsetAsync(sum, 0, sizeof(float) * 16, stream);
    edge_sig_deg<<<(E + 255) / 256, 256, 0, stream>>>(ea, dst, dt_w, dt_b, ew, deg, sum, E);
    dinv_kernel<<<(n + 255) / 256, 256, 0, stream>>>(deg, sum, dinv, n);

    // ---- layer 1: GEMM -> scatter -> self-loop + bias + LN + ReLU ---------
    gemm128_wmma<<<gemmBlocks, 128, 0, stream>>>(x, W1, bufA, n);
    hipMemsetAsync(bufB, 0, sizeof(float) * (size_t)n * FD, stream);
    scatter_edges<<<edgeBlocks, 256, 0, stream>>>(bufA, ew, sum, dinv, src, dst, bufB, E);
    finish_ln<true><<<rowBlocks, 256, 0, stream>>>(bufB, bufA, dinv, b1, g1, be1, bufA, n);

    // ---- layer 2: GEMM -> scatter -> self-loop + bias + LN ----------------
    gemm128_wmma<<<gemmBlocks, 128, 0, stream>>>(bufA, W2, bufB, n);
    hipMemsetAsync(out, 0, sizeof(float) * (size_t)n * FD, stream);
    scatter_edges<<<edgeBlocks, 256, 0, stream>>>(bufB, ew, sum, dinv, src, dst, out, E);
    finish_ln<false><<<rowBlocks, 256, 0, stream>>>(out, bufB, dinv, b2, g2, be2, out, n);
}